// RingCornerNet_87531433492878
// MI455X (gfx1250) — compile-verified
//
#include <hip/hip_runtime.h>
#include <stdint.h>

#define B_N 2048
#define R_N 256
#define D_N 256
#define TB 16
#define TR 16
#define CH 64
#define NCH (D_N / CH)
#define XPAD 68           // x row stride (floats): (68*bl+d)%64=(4*bl+d)%64 conflict-free, rows 16B-aligned
#define PROW (CH * 6)     // 384 floats: packed params per ring per chunk

typedef float v2f __attribute__((ext_vector_type(2)));
typedef float v8f __attribute__((ext_vector_type(8)));

__device__ __forceinline__ void async_b128(uint32_t lds_off, const void* gptr) {
  asm volatile("global_load_async_to_lds_b128 %0, %1, off"
               :: "v"(lds_off), "v"(gptr) : "memory");
}
__device__ __forceinline__ void wait_async0() {
  asm volatile("s_wait_asynccnt 0" ::: "memory");
}
__device__ __forceinline__ uint32_t lds_off(const void* p) {
  return (uint32_t)(uintptr_t)p;
}

// d_ws layout:
//   WP   : R * 128 pairs * 12 floats (per d-pair: ks2_o,c2_o,w_o,ks2_i,c2_i,w_i for d0 then d1)
//          where ks2 = -log2(e)*kappa*sign, c2 = -ks2*th  =>  u = exp2(ks2*x + c2) = e^{-a}
//   ypart: B * 16 floats (partial head sums per ring-block)
#define WP_FLOATS ((size_t)R_N * 128 * 12)

__global__ void __launch_bounds__(256)
prep_kernel(const float* __restrict__ th_o, const float* __restrict__ sign_o,
            const float* __restrict__ mask_o, const float* __restrict__ th_i,
            const float* __restrict__ sign_i, const float* __restrict__ mask_i,
            const float* __restrict__ log_kappa, float* __restrict__ WP) {
  int p = blockIdx.x * blockDim.x + threadIdx.x;  // r*128 + dpair
  if (p >= R_N * 128) return;
  int r = p >> 7, dp = p & 127;
  int idx = r * D_N + dp * 2;
  const float L2E = 1.4426950408889634f;
  float kn = -L2E * __expf(log_kappa[0]);

  float ks0o = kn * sign_o[idx];     float c0o = -ks0o * th_o[idx];
  float w0o  = 1.f / (1.f + __expf(mask_o[idx]));          // 1 - sigmoid(mask)
  float ks0i = kn * sign_i[idx];     float c0i = -ks0i * th_i[idx];
  float w0i  = 1.f / (1.f + __expf(mask_i[idx]));
  float ks1o = kn * sign_o[idx + 1]; float c1o = -ks1o * th_o[idx + 1];
  float w1o  = 1.f / (1.f + __expf(mask_o[idx + 1]));
  float ks1i = kn * sign_i[idx + 1]; float c1i = -ks1i * th_i[idx + 1];
  float w1i  = 1.f / (1.f + __expf(mask_i[idx + 1]));

  float* q = WP + (size_t)p * 12;
  float4 a = {ks0o, c0o, w0o, ks0i};
  float4 b = {c0i, w0i, ks1o, c1o};
  float4 c = {w1o, ks1i, c1i, w1i};
  *(float4*)(q)     = a;
  *(float4*)(q + 4) = b;
  *(float4*)(q + 8) = c;
}

// one sigmoid-blend term: pn *= (1 + w*u), pd *= (1+u), u = exp2(ks2*x + c2)
#define TERM(KS, C, W, XV, PN, PD)                              \
  {                                                             \
    float u = __builtin_amdgcn_exp2f(fmaf((KS), (XV), (C)));    \
    (PN) = (PN) * fmaf((W), u, 1.f);                            \
    (PD) = fmaf((PD), u, (PD));                                 \
  }

__global__ void __launch_bounds__(256)
ring_kernel(const float* __restrict__ x, const float* __restrict__ WP,
            const float* __restrict__ head_w, float* __restrict__ ypart) {
  __shared__ __align__(16) float xs[2][TB * XPAD];
  __shared__ __align__(16) float ps[2][TR * PROW];
  __shared__ __align__(16) float zt[TR * TB];

  const int t   = threadIdx.x;
  const int bl  = t & 15;
  const int rl  = t >> 4;
  const int row = t >> 4;     // staging row (0..15)
  const int l16 = t & 15;     // staging lane-in-row
  const int b0  = blockIdx.x * TB;
  const int r0  = blockIdx.y * TR;

  auto stage = [&](int buf, int ch) {
    const int d0  = ch * CH;
    const int dp0 = ch * (CH / 2);
    // params: 16 rings * 384 floats = 1536 B128 transfers, 6 per thread
    const float* srow = WP + ((size_t)(r0 + row) * 128 + dp0) * 12;
    uint32_t drow = lds_off(&ps[buf][row * PROW]);
#pragma unroll
    for (int it = 0; it < 6; ++it) {
      int k = l16 + 16 * it;                 // B128 index within row (0..95)
      async_b128(drow + (uint32_t)k * 16, srow + k * 4);
    }
    // x: 16 rows * 64 floats = 256 B128 transfers, 1 per thread
    const float* sx = x + (size_t)(b0 + row) * D_N + d0 + l16 * 4;
    async_b128(lds_off(&xs[buf][row * XPAD + l16 * 4]), sx);
  };

  stage(0, 0);
  wait_async0();
  __syncthreads();

  // running products, numerator normalized by rcp(denominator) every 4 dims
  float pn_o = 1.f, pd_o = 1.f, pn_i = 1.f, pd_i = 1.f;
  int buf = 0;
  for (int ch = 0; ch < NCH; ++ch) {
    if (ch + 1 < NCH) stage(buf ^ 1, ch + 1);
    const float* pr = &ps[buf][rl * PROW];
    const float* xr = &xs[buf][bl * XPAD];
#pragma unroll
    for (int g = 0; g < CH / 4; ++g) {        // group of 4 dims = 2 packed pairs
      float4 xv = *(const float4*)(xr + g * 4);
      const float* pb = pr + g * 24;
      float4 q0 = *(const float4*)(pb);
      float4 q1 = *(const float4*)(pb + 4);
      float4 q2 = *(const float4*)(pb + 8);
      float4 q3 = *(const float4*)(pb + 12);
      float4 q4 = *(const float4*)(pb + 16);
      float4 q5 = *(const float4*)(pb + 20);
      // d = 4g+0
      TERM(q0.x, q0.y, q0.z, xv.x, pn_o, pd_o)
      TERM(q0.w, q1.x, q1.y, xv.x, pn_i, pd_i)
      // d = 4g+1
      TERM(q1.z, q1.w, q2.x, xv.y, pn_o, pd_o)
      TERM(q2.y, q2.z, q2.w, xv.y, pn_i, pd_i)
      // d = 4g+2
      TERM(q3.x, q3.y, q3.z, xv.z, pn_o, pd_o)
      TERM(q3.w, q4.x, q4.y, xv.z, pn_i, pd_i)
      // d = 4g+3
      TERM(q4.z, q4.w, q5.x, xv.w, pn_o, pd_o)
      TERM(q5.y, q5.z, q5.w, xv.w, pn_i, pd_i)
      // normalize (u <= ~2^20 for these inputs => pd <= ~2^84, safe in f32)
      pn_o *= __builtin_amdgcn_rcpf(pd_o); pd_o = 1.f;
      pn_i *= __builtin_amdgcn_rcpf(pd_i); pd_i = 1.f;
    }
    wait_async0();
    __syncthreads();
    buf ^= 1;
  }

  // z tile: zt[k*16 + m]  (k = local ring, m = local batch) -> WMMA A layout
  zt[rl * TB + bl] = pn_o * (1.f - pn_i);
  __syncthreads();

  // Head partial GEMV via f32 WMMA on wave 0 (EXEC all-1s within the wave).
  // A = z (16x16 over K-chunks of 4), B[k][n] = w[k] replicated across N,
  // so every column of D holds y_partial[m].
  if (t < 32) {
    const int lane = t;
    const int half = lane >> 4;
    const int L    = lane & 15;
    const float* wrow = head_w + r0;
    v8f acc = {0.f, 0.f, 0.f, 0.f, 0.f, 0.f, 0.f, 0.f};
#pragma unroll
    for (int k0 = 0; k0 < 16; k0 += 4) {
      int kA = k0 + 2 * half;  // lanes 0-15: K={k0,k0+1}; lanes 16-31: K={k0+2,k0+3}
      v2f A;  A.x = zt[kA * TB + L];      A.y = zt[(kA + 1) * TB + L];
      v2f Bv; Bv.x = wrow[kA];            Bv.y = wrow[kA + 1];
      acc = __builtin_amdgcn_wmma_f32_16x16x4_f32(false, A, false, Bv,
                                                  (short)0, acc, false, false);
    }
    // D layout: VGPR i, lane 0 -> M=i; lane 16 -> M=8+i  (N=0 column)
    if (L == 0) {
#pragma unroll
      for (int i = 0; i < 8; ++i)
        ypart[(size_t)(b0 + half * 8 + i) * 16 + blockIdx.y] = acc[i];
    }
  }
}

__global__ void __launch_bounds__(256)
reduce_kernel(const float* __restrict__ ypart, const float* __restrict__ head_b,
              float* __restrict__ out) {
  int b = blockIdx.x * blockDim.x + threadIdx.x;
  if (b >= B_N) return;
  float s = head_b[0];
#pragma unroll
  for (int j = 0; j < 16; ++j) s += ypart[(size_t)b * 16 + j];
  out[b] = s;
}

extern "C" void kernel_launch(void* const* d_in, const int* in_sizes, int n_in,
                              void* d_out, int out_size, void* d_ws, size_t ws_size,
                              hipStream_t stream) {
  const float* x         = (const float*)d_in[0];
  const float* th_o      = (const float*)d_in[1];
  const float* sign_o    = (const float*)d_in[2];
  const float* mask_o    = (const float*)d_in[3];
  const float* th_i      = (const float*)d_in[4];
  const float* sign_i    = (const float*)d_in[5];
  const float* mask_i    = (const float*)d_in[6];
  const float* log_kappa = (const float*)d_in[7];
  const float* head_w    = (const float*)d_in[8];
  const float* head_b    = (const float*)d_in[9];

  float* WP    = (float*)d_ws;
  float* ypart = WP + WP_FLOATS;
  float* out   = (float*)d_out;

  prep_kernel<<<(R_N * 128) / 256, 256, 0, stream>>>(th_o, sign_o, mask_o,
                                                     th_i, sign_i, mask_i,
                                                     log_kappa, WP);
  dim3 grid(B_N / TB, R_N / TR);
  ring_kernel<<<grid, 256, 0, stream>>>(x, WP, head_w, ypart);
  reduce_kernel<<<(B_N + 255) / 256, 256, 0, stream>>>(ypart, head_b, out);
}